// VariationalBayesRouting2D_335007449617
// MI455X (gfx1250) — compile-verified
//
#include <hip/hip_runtime.h>
#include <cstdint>

// ---------------------------------------------------------------------------
// VariationalBayesRouting2D for MI455X (gfx1250, wave32).
//   BATCH=4, BCAPS=32, CCAPS=32, D=4, K=3, FO=14, FI=27, N_ITER=3
// Design notes (v2):
//   * 256-block decomposition (b,{i|j},o-half) to cover the large WGP array.
//   * Double-buffered CDNA5 async global->LDS staging
//     (global_load_async_to_lds_b64 + s_wait_asynccnt): prefetch of stage t+1
//     overlaps compute of stage t; wait-0 before the barrier makes all copies
//     visible block-wide.
//   * 32-wide output-capsule sum (E-step denominator) on the matrix pipe via
//     V_WMMA_F32_16X16X4_F32 with a ones A-matrix (cross-lane reduction).
//   * Fully deterministic: no atomics anywhere; fold/unfold rebuilt by gather
//     (k_grid); BN statistics via per-block partials summed in fixed order.
// Working set (~150 MB incl. V=116 MB) fits the 192 MB L2, so the 5 logical
// passes over V run from L2; E-step cost is ~14.5M v_exp_f32 on the trans pipe.
// Workspace requirement: ~32 MB (layout in kernel_launch).
// ---------------------------------------------------------------------------

#define NB 4
#define NCAP 32
#define ND 4
#define NO 196          // 14*14
#define NOH 98          // o-half
#define NKK 9           // 3*3
#define NCELL 1764      // NO*NKK
#define NCELLH 882      // NOH*NKK
#define NJO 25088       // NB*NCAP*NO
#define GRID_PAD 736    // 27*27=729 padded
#define LN2PI_F 1.8378770664093453f

typedef float v2f __attribute__((ext_vector_type(2)));
typedef float v8f __attribute__((ext_vector_type(8)));

#if defined(__has_builtin)
#if __has_builtin(__builtin_amdgcn_wmma_f32_16x16x4_f32)
#define HAVE_WMMA_F32X4 1
#endif
#endif

// ---- CDNA5 async global->LDS copy, 8-byte granules (slabs are 8B aligned) --
__device__ __forceinline__ void async_b64(float* ldsDst, const float* g, int nfloats){
  uint32_t lbase = (uint32_t)(uintptr_t)ldsDst;   // low 32b of generic ptr == LDS offset
  int nv = nfloats >> 1;
  for (int v = (int)threadIdx.x; v < nv; v += (int)blockDim.x){
    uint32_t la = lbase + ((uint32_t)v << 3);
    const float* ga = g + ((size_t)v << 1);
    asm volatile("global_load_async_to_lds_b64 %0, %1, off"
                 :: "v"(la), "v"(ga) : "memory");
  }
}
__device__ __forceinline__ void wait_async_all(){
  asm volatile("s_wait_asynccnt 0" ::: "memory");
}

// ---- digamma for x >= 1 (recurrence + asymptotic series) ----
__device__ __forceinline__ float digammaf_(float x){
  float r = 0.f;
  while (x < 6.f){ r -= 1.f / x; x += 1.f; }
  float inv = 1.f / x, inv2 = inv * inv;
  return r + __logf(x) - 0.5f*inv
         - inv2*(1.f/12.f - inv2*(1.f/120.f - inv2*(1.f/252.f)));
}

// ---------------------------------------------------------------------------
// M-step: per (b,j,o) weighted statistics over (i,k) [288 terms].
// Block = (b, j, o-half); 128 threads (98 compute); i-loop double-buffered.
// ---------------------------------------------------------------------------
#define KS_A 0          // 882
#define KS_P 882        // 882
#define KS_G 1764       // 736
#define KS_V 2500       // 3528 ([d][ol][k])
#define KS_BUF 6028
__global__ void k_stats(const float* __restrict__ aI, const float* __restrict__ V,
                        const float* __restrict__ P, const float* __restrict__ gridG,
                        float* __restrict__ Rj, float* __restrict__ mArr,
                        float* __restrict__ invPsiA, float* __restrict__ lndetA, int it)
{
  extern __shared__ float sm[];
  const int tid  = (int)threadIdx.x;
  const int bj   = (int)blockIdx.x >> 1;
  const int half = (int)blockIdx.x & 1;
  const int b = bj >> 5, j = bj & 31;

  auto stage = [&](int i, float* B){
    size_t bi = (size_t)(b*NCAP + i);
    async_b64(B + KS_A, aI + bi*NCELL + half*NCELLH, NCELLH);
    if (it > 0){
      async_b64(B + KS_P, P + (bi*NCAP + j)*NCELL + half*NCELLH, NCELLH);
      async_b64(B + KS_G, gridG + bi*GRID_PAD, GRID_PAD);
    }
    for (int d = 0; d < ND; ++d)
      async_b64(B + KS_V + d*NCELLH,
                V + (bi*NCAP + j)*((size_t)ND*NCELL) + (size_t)d*NCELL + half*NCELLH,
                NCELLH);
  };

  const int ol = tid;                  // local o (0..97 active)
  const int o  = half*NOH + ol;
  const int oy = o / 14, ox = o % 14;

  float r = 0.f;
  float s1[ND] = {0.f,0.f,0.f,0.f};
  float s2[ND] = {0.f,0.f,0.f,0.f};

  stage(0, sm);                        // prefetch i=0 into buffer 0
  for (int i = 0; i < NCAP; ++i){
    float* B = sm + (i & 1)*KS_BUF;
    wait_async_all();                  // my copies for stage(i) landed
    __syncthreads();                   // => everyone's copies landed
    if (i + 1 < NCAP) stage(i + 1, sm + ((i + 1) & 1)*KS_BUF);  // overlap

    if (ol < NOH){
      #pragma unroll
      for (int k = 0; k < NKK; ++k){
        int ky = k / 3, kx = k - ky*3;
        float av = B[KS_A + ol*NKK + k];
        float rw;
        if (it == 0){
          rw = av * (1.f/32.f);
        } else {
          float den = fmaxf(B[KS_G + (oy*2+ky)*27 + (ox*2+kx)], 1e-11f);
          rw = (B[KS_P + ol*NKK + k] / den) * av;
        }
        r += rw;
        #pragma unroll
        for (int d = 0; d < ND; ++d){
          float vv = B[KS_V + d*NCELLH + ol*NKK + k];
          s1[d] += rw * vv;
          s2[d] += rw * vv * vv;
        }
      }
    }
  }

  if (ol < NOH){
    int sjo = (b*NCAP + j)*NO + o;
    float R   = r;
    float kap = 1.f + R;                       // KAPPA0 + R_j
    float ld  = 0.f;
    #pragma unroll
    for (int d = 0; d < ND; ++d){
      float mu  = s1[d] / R;
      float sig = s2[d] - R*mu*mu;             // sum Rw (V-mu)^2
      float ip  = 0.05f + sig + (R/kap)*mu*mu; // PSI0 + ...
      invPsiA[(size_t)sjo*ND + d] = ip;
      mArr[(size_t)sjo*ND + d]    = (R/kap)*mu;  // m_j
      ld -= __logf(ip);
    }
    Rj[sjo]     = R;
    lndetA[sjo] = ld;
  }
}

// ---- alphasum per (b,o): digamma(sum_j alpha_j), fixed order ----
__global__ void k_prep1(const float* __restrict__ Rj, float* __restrict__ dgA){
  int t = (int)blockIdx.x*(int)blockDim.x + (int)threadIdx.x;
  if (t >= NB*NO) return;
  int b = t / NO, o = t % NO;
  float s = 0.f;
  for (int j = 0; j < NCAP; ++j) s += Rj[(size_t)(b*NCAP + j)*NO + o];
  dgA[t] = digammaf_(32.f + s);   // CCAPS*ALPHA0 + sum R_j
}

// ---- per (b,j,o): base = Elnpi + .5*Elam - 2ln2pi - 2/kappa ; w = .5*nu/invPsi
__global__ void k_prep2(const float* __restrict__ Rj, const float* __restrict__ lndetA,
                        const float* __restrict__ invPsiA, const float* __restrict__ dgA,
                        float* __restrict__ baseA, float* __restrict__ wA)
{
  int t = (int)blockIdx.x*(int)blockDim.x + (int)threadIdx.x;
  if (t >= NJO) return;
  int b = t / (NCAP*NO);
  int o = t % NO;
  float R     = Rj[t];
  float alpha = 1.f + R, kap = 1.f + R, nu = 6.f + R;  // NU0 = D+2 = 6
  float Elnpi = digammaf_(alpha) - dgA[b*NO + o];
  float lg = 0.f;
  #pragma unroll
  for (int d = 0; d < ND; ++d) lg += lgammaf(0.5f*(nu - (float)d));
  float Elam = 2.f*__logf(nu) + 2.f - lg + lndetA[t];
  baseA[t] = Elnpi + 0.5f*Elam - 2.f*LN2PI_F - 2.f/kap;
  #pragma unroll
  for (int d = 0; d < ND; ++d) wA[(size_t)t*ND + d] = 0.5f*nu / invPsiA[(size_t)t*ND + d];
}

// ---------------------------------------------------------------------------
// E-step: p = exp(base - sum_d w_d (V_d - m_d)^2); stores p and per-cell
// sum_j p. Block = (b,i,o-half), 128 threads; 16 double-buffered groups of 2
// output capsules; wave = 16 cells x (2 j across lane halves); j-sum on the
// matrix pipe (WMMA f32 16x16x4, A = ones).
// ---------------------------------------------------------------------------
#define KE_CS   0        // 882 : per-cell sum_j p accumulator
#define KE_BASE 0        // within buffer: 196 (2j x 98)
#define KE_M    196      // 784 (2j x 98 x 4)
#define KE_W    980      // 784
#define KE_V    1764     // 7056 (2j x 4d x 98 x 9)
#define KE_BUF  8820
__global__ void k_estep(const float* __restrict__ V, const float* __restrict__ baseA,
                        const float* __restrict__ mA, const float* __restrict__ wA,
                        float* __restrict__ P, float* __restrict__ cellSumG)
{
  extern __shared__ float sm[];
  float* cellSumL = sm;                      // 882
  float* bufs     = sm + NCELLH;             // 2 x 8820
  const int tid  = (int)threadIdx.x;
  const int bi   = (int)blockIdx.x >> 1;
  const int half = (int)blockIdx.x & 1;
  const int b = bi >> 5, i = bi & 31;
  const int lane = tid & 31, wave = tid >> 5;
  const int h = lane >> 4, c16 = lane & 15;

  auto stage = [&](int g, float* B){
    for (int jl = 0; jl < 2; ++jl){
      size_t jj = (size_t)(b*NCAP + 2*g + jl);
      async_b64(B + KE_BASE + jl*NOH,    baseA + jj*NO + half*NOH, NOH);
      async_b64(B + KE_M    + jl*NOH*ND, mA + (jj*NO + half*NOH)*ND, NOH*ND);
      async_b64(B + KE_W    + jl*NOH*ND, wA + (jj*NO + half*NOH)*ND, NOH*ND);
      for (int d = 0; d < ND; ++d)
        async_b64(B + KE_V + (jl*ND + d)*NCELLH,
                  V + ((size_t)(b*NCAP + i)*NCAP + 2*g + jl)*((size_t)ND*NCELL)
                    + (size_t)d*NCELL + half*NCELLH,
                  NCELLH);
    }
  };

  stage(0, bufs);                            // prefetch group 0
  for (int t = tid; t < NCELLH; t += 128) cellSumL[t] = 0.f;

  v2f onesA; onesA.x = 1.f; onesA.y = 1.f;

  for (int g = 0; g < 16; ++g){              // 16 groups of 2 output capsules
    float* B = bufs + (g & 1)*KE_BUF;
    wait_async_all();
    __syncthreads();                         // stage(g) visible; prior readers done
    if (g + 1 < 16) stage(g + 1, bufs + ((g + 1) & 1)*KE_BUF);  // overlap

    for (int cb = wave*16; cb < NCELLH; cb += 64){
      int cellL = cb + c16;
      bool valid = cellL < NCELLH;
      int cl = valid ? cellL : (NCELLH - 1);
      int ol = cl / NKK, k = cl - ol*NKK;
      int o  = half*NOH + ol;
      int jl = h;                            // local j in [0,2)
      float q = B[KE_BASE + jl*NOH + ol];
      #pragma unroll
      for (int d = 0; d < ND; ++d){
        float vv = B[KE_V + (jl*ND + d)*NCELLH + ol*NKK + k];
        float dm = vv - B[KE_M + (jl*NOH + ol)*ND + d];
        q -= B[KE_W + (jl*NOH + ol)*ND + d] * dm * dm;
      }
      float p = __expf(q);
      if (valid)
        P[((size_t)((b*NCAP + i)*NCAP + 2*g + h)*NO + o)*NKK + k] = p;

      v2f bm; bm.x = valid ? p : 0.f; bm.y = 0.f;
      float colsum;
#ifdef HAVE_WMMA_F32X4
      // D = ones(16x4) x B(4x16): every row holds the per-column sum of the
      // live lane-half slots -> sum over this group's 2 j's; accumulated over
      // 16 groups via cellSumL (unique writer, no atomics).
      v8f cz = {};
      v8f dm_ = __builtin_amdgcn_wmma_f32_16x16x4_f32(
          false, onesA, false, bm, (short)0, cz, false, false);
      colsum = dm_[0];
#else
      colsum = bm.x + __shfl_xor(bm.x, 16, 32);
#endif
      if (h == 0 && valid)
        cellSumL[cellL] += colsum;
    }
  }
  __syncthreads();
  for (int t = tid; t < NCELLH; t += 128)
    cellSumG[(size_t)bi*NCELL + half*NCELLH + t] = cellSumL[t];
}

// ---- fold/unfold grid: gather <=4 contributing cells, fixed order ----
__global__ void k_grid(const float* __restrict__ cellSumG, float* __restrict__ gridG){
  int t = (int)blockIdx.x*(int)blockDim.x + (int)threadIdx.x;
  if (t >= NB*NCAP*729) return;
  int bi = t / 729, yx = t % 729;
  int y = yx / 27, x = yx % 27;
  float acc = 0.f;
  #pragma unroll
  for (int ky = 0; ky < 3; ++ky){
    int ry = y - ky;
    if (ry < 0 || (ry & 1)) continue;
    int oy = ry >> 1; if (oy >= 14) continue;
    #pragma unroll
    for (int kx = 0; kx < 3; ++kx){
      int rx = x - kx;
      if (rx < 0 || (rx & 1)) continue;
      int ox = rx >> 1; if (ox >= 14) continue;
      acc += cellSumG[(size_t)bi*NCELL + (oy*14 + ox)*NKK + ky*3 + kx];
    }
  }
  gridG[(size_t)bi*GRID_PAD + yx] = acc;
}

// ---- deterministic block-sum helper (wave32 shuffles + LDS) ----
__device__ __forceinline__ float blockSum256(float v, float* sb){
  #pragma unroll
  for (int ofs = 16; ofs > 0; ofs >>= 1) v += __shfl_down(v, ofs, 32);
  int lane = (int)threadIdx.x & 31, w = (int)threadIdx.x >> 5;
  __syncthreads();
  if (lane == 0) sb[w] = v;
  __syncthreads();
  if (w == 0){
    v = (lane < 8) ? sb[lane] : 0.f;
    v += __shfl_down(v, 4, 32);
    v += __shfl_down(v, 2, 32);
    v += __shfl_down(v, 1, 32);
  }
  return v;  // valid in thread 0
}

// ---- final per-(b,j,o) activations + per-block BN partials ----
__global__ void k_final1(const float* __restrict__ Rj, const float* __restrict__ lndetA,
                         const float* __restrict__ mA, const float* __restrict__ dgA,
                         float* __restrict__ xa, float* __restrict__ partial)
{
  __shared__ float sb[8];
  int t = (int)blockIdx.x*256 + (int)threadIdx.x;   // grid sized exactly: t < NJO
  int b = t / (NCAP*NO);
  int o = t % NO;
  float R = Rj[t];
  float alpha = 1.f + R, nu = 6.f + R;
  float Elnpi = digammaf_(alpha) - dgA[b*NO + o];
  float lg = 0.f;
  #pragma unroll
  for (int d = 0; d < ND; ++d) lg += lgammaf(0.5f*(nu - (float)d));
  float Elam = 2.f*__logf(nu) + 2.f - lg + lndetA[t];
  float Hq = 2.f*LN2PI_F - 0.5f*Elam;
  float x  = -__expf(Elnpi) * Hq;
  xa[t] = x;

  float smv = 0.f, sqv = 0.f;
  #pragma unroll
  for (int d = 0; d < ND; ++d){ float m = mA[(size_t)t*ND + d]; smv += m; sqv += m*m; }

  float v0 = blockSum256(x,   sb);
  float v1 = blockSum256(x*x, sb);
  float v2 = blockSum256(smv, sb);
  float v3 = blockSum256(sqv, sb);
  if (threadIdx.x == 0){
    partial[(size_t)blockIdx.x*4 + 0] = v0;
    partial[(size_t)blockIdx.x*4 + 1] = v1;
    partial[(size_t)blockIdx.x*4 + 2] = v2;
    partial[(size_t)blockIdx.x*4 + 3] = v3;
  }
}

__global__ void k_bnsum(const float* __restrict__ partial, float* __restrict__ bnAcc){
  int q = (int)threadIdx.x;
  if (q < 4){
    float s = 0.f;
    for (int blk = 0; blk < 98; ++blk) s += partial[(size_t)blk*4 + q];
    bnAcc[q] = s;
  }
}

// ---- BN + sigmoid; outputs: a_j (25088) then m_out (100352) ----
__global__ void k_final2(const float* __restrict__ xa, const float* __restrict__ mA,
                         const float* __restrict__ bnAcc,
                         const float* __restrict__ gA, const float* __restrict__ bA,
                         const float* __restrict__ gV, const float* __restrict__ bV,
                         float* __restrict__ out)
{
  int t = (int)blockIdx.x*(int)blockDim.x + (int)threadIdx.x;
  const int NA = NJO, NM = NJO*ND;
  if (t >= NA + NM) return;
  if (t < NA){
    float mean = bnAcc[0] * (1.f/(float)NA);
    float var  = bnAcc[1] * (1.f/(float)NA) - mean*mean;
    float xh = (xa[t] - mean) * rsqrtf(var + 1e-3f) * gA[0] + bA[0];
    out[t] = 1.f / (1.f + __expf(-xh));
  } else {
    int tm = t - NA;                 // flat (b*32+j, d, o)
    int bj = tm / (ND*NO);
    int rest = tm % (ND*NO);
    int d = rest / NO, o = rest % NO;
    float mean = bnAcc[2] * (1.f/(float)NM);
    float var  = bnAcc[3] * (1.f/(float)NM) - mean*mean;
    float m = mA[((size_t)bj*NO + o)*ND + d];
    out[t] = (m - mean) * rsqrtf(var + 1e-3f) * gV[0] + bV[0];
  }
}

// ---------------------------------------------------------------------------
extern "C" void kernel_launch(void* const* d_in, const int* in_sizes, int n_in,
                              void* d_out, int out_size, void* d_ws, size_t ws_size,
                              hipStream_t stream) {
  const float* aI = (const float*)d_in[0];
  const float* V  = (const float*)d_in[1];
  const float* gA = (const float*)d_in[2];
  const float* bA = (const float*)d_in[3];
  const float* gV = (const float*)d_in[4];
  const float* bV = (const float*)d_in[5];
  float* out = (float*)d_out;

  // Workspace layout (floats). Total ~7.95M floats (~32 MB); all async-staged
  // slabs are >=8B aligned.
  float* ws       = (float*)d_ws;
  float* P        = ws;                      // 7,225,344
  float* Rj       = P        + 7225344;      //    25,088
  float* mArr     = Rj       + 25088;        //   100,352
  float* invPsiA  = mArr     + 100352;       //   100,352
  float* lndetA   = invPsiA  + 100352;       //    25,088
  float* baseA    = lndetA   + 25088;        //    25,088
  float* wA       = baseA    + 25088;        //   100,352
  float* dgA      = wA       + 100352;       //       784
  float* gridG    = dgA      + 784;          //    94,208
  float* cellSumG = gridG    + 94208;        //   225,792
  float* xa       = cellSumG + 225792;       //    25,088
  float* partial  = xa       + 25088;        //       392
  float* bnAcc    = partial  + 392;          //         4

  const size_t shStats = (size_t)(2*KS_BUF) * sizeof(float);            // 48,224 B
  const size_t shEstep = (size_t)(NCELLH + 2*KE_BUF) * sizeof(float);   // 74,088 B

  for (int it = 0; it < 3; ++it){
    k_stats<<<256, 128, shStats, stream>>>(aI, V, P, gridG, Rj, mArr, invPsiA, lndetA, it);
    k_prep1<<<4, 256, 0, stream>>>(Rj, dgA);
    if (it < 2){
      k_prep2<<<98, 256, 0, stream>>>(Rj, lndetA, invPsiA, dgA, baseA, wA);
      k_estep<<<256, 128, shEstep, stream>>>(V, baseA, mArr, wA, P, cellSumG);
      k_grid<<<365, 256, 0, stream>>>(cellSumG, gridG);
    }
  }
  k_final1<<<98, 256, 0, stream>>>(Rj, lndetA, mArr, dgA, xa, partial);
  k_bnsum<<<1, 32, 0, stream>>>(partial, bnAcc);
  k_final2<<<490, 256, 0, stream>>>(xa, mArr, bnAcc, gA, bA, gV, bV, out);
}